// Model_27144193311108
// MI455X (gfx1250) — compile-verified
//
#include <hip/hip_runtime.h>
#include <stdint.h>

typedef __attribute__((ext_vector_type(2))) float v2f;
typedef __attribute__((ext_vector_type(8))) float v8f;

#define HIDC 128      // hidden width (reference: 128)
#define HT   8        // HIDC / 16
#define FINC 256      // input feature width (reference: 256)
#define KCH  64       // K-chunk staged in LDS per iteration
#define LROW 66       // KCH + 2 floats pad -> row stride 264B, kills bank conflicts

// ---------------------------------------------------------------------------
// Native fp32 global atomic add (GLOBAL_ATOMIC_ADD_F32, no-return form).
// ---------------------------------------------------------------------------
__device__ __forceinline__ void atomic_fadd(float* p, float v) {
    asm volatile("global_atomic_add_f32 %0, %1, off"
                 :
                 : "v"((unsigned long long)(uintptr_t)p), "v"(v)
                 : "memory");
}

// ---------------------------------------------------------------------------
// CDNA5 async global->LDS copy (ASYNCcnt-tracked) + its drain.
// dsaddr comes from the low 32 bits of the flat LDS pointer (ISA 10.2:
// LDS aperture maps addr[31:0] directly to the LDS address).
// ---------------------------------------------------------------------------
__device__ __forceinline__ void async_g2l_b128(float* lds_dst, const float* g_src) {
    asm volatile("global_load_async_to_lds_b128 %0, %1, off"
                 :
                 : "v"((uint32_t)(uintptr_t)lds_dst),
                   "v"((unsigned long long)(uintptr_t)g_src)
                 : "memory");
}
__device__ __forceinline__ void wait_async0() {
    asm volatile("s_wait_asynccnt 0x0" ::: "memory");
}

// ---------------------------------------------------------------------------
// Degree / norm kernels
// ---------------------------------------------------------------------------
__global__ void fill_u32(unsigned* p, unsigned v, int n) {
    int i = blockIdx.x * blockDim.x + threadIdx.x;
    if (i < n) p[i] = v;
}

__global__ void deg_count(unsigned* deg, const int* __restrict__ col, int nE) {
    int e = blockIdx.x * blockDim.x + threadIdx.x;
    if (e < nE) atomicAdd(&deg[col[e]], 1u);
}

__global__ void deg_to_dinv(float* __restrict__ dinv, const unsigned* __restrict__ deg, int n) {
    int i = blockIdx.x * blockDim.x + threadIdx.x;
    if (i < n) dinv[i] = rsqrtf((float)deg[i]);   // deg >= 1 (self loop)
}

// ---------------------------------------------------------------------------
// Dense:  out[M][128] = x[M][256] @ W^T + bias,  via V_WMMA_F32_16X16X4_F32.
// Block = 256 threads = 8 waves; each wave owns one 16-row M-tile and all
// 128 output columns (8 v8f accumulators). W is staged into LDS in 64-wide
// K-chunks with async global->LDS loads shared by all 8 waves.
//   A frag (16x4):  vgpr v -> K = 2*(lane>>4)+v,  M = lane&15
//   B frag (4x16):  vgpr v -> K = 2*(lane>>4)+v,  N = lane&15   (B = W^T)
//   D (16x16):      vgpr v -> M = 8*(lane>>4)+v,  N = lane&15
// ---------------------------------------------------------------------------
__global__ void gemm_xWt_bias(const float* __restrict__ x,
                              const float* __restrict__ W,
                              const float* __restrict__ bias,
                              float* __restrict__ out,
                              int mTiles) {
    __shared__ float wlds[HIDC * LROW];            // 33792 B

    int tid  = threadIdx.x;
    int wave = tid >> 5;
    int lane = tid & 31;
    int tile = blockIdx.x * 8 + wave;
    bool active = tile < mTiles;
    int tileC = active ? tile : (mTiles - 1);      // clamp: keep all waves in barriers

    int r16 = lane & 15;
    int kh  = lane >> 4;

    const float* xr = x + (size_t)(tileC * 16 + r16) * FINC + 2 * kh;

    v8f acc[HT] = {};

    for (int kc = 0; kc < FINC; kc += KCH) {
        __syncthreads();                           // previous chunk fully consumed
        // stage W[:, kc:kc+64]: 128 rows x 16 float4 = 2048 xfers, 8 per thread
#pragma unroll
        for (int i = 0; i < 8; ++i) {
            int idx = tid + i * 256;
            int row = idx >> 4;
            int q   = idx & 15;
            async_g2l_b128(&wlds[row * LROW + q * 4],
                           W + (size_t)row * FINC + kc + q * 4);
        }
        wait_async0();
        __syncthreads();                           // chunk visible to all waves

        for (int kl = 0; kl < KCH; kl += 4) {
            v2f a = *(const v2f*)(xr + kc + kl);
            v2f bf[HT];
#pragma unroll
            for (int nt = 0; nt < HT; ++nt)        // batch all B loads (LDS)
                bf[nt] = *(const v2f*)(&wlds[(nt * 16 + r16) * LROW + kl + 2 * kh]);
#pragma unroll
            for (int nt = 0; nt < HT; ++nt)        // then the WMMA burst
                acc[nt] = __builtin_amdgcn_wmma_f32_16x16x4_f32(
                    false, a, false, bf[nt], (short)0, acc[nt], false, false);
        }
    }

    if (active) {
        int mBase = tile * 16 + 8 * kh;
#pragma unroll
        for (int nt = 0; nt < HT; ++nt) {
            float bb = bias[nt * 16 + r16];
#pragma unroll
            for (int v = 0; v < 8; ++v) {
                out[(size_t)(mBase + v) * HIDC + nt * 16 + r16] = acc[nt][v] + bb;
            }
        }
    }
}

// ---------------------------------------------------------------------------
// SpMM pass, step 1: self-loop term (fully initializes T):
//   T[i][:] = (base ? base[i][:] : 0) + dinv[i]^2 * S[i][:]
// ---------------------------------------------------------------------------
__global__ void spmm_init(float* __restrict__ T, const float* __restrict__ S,
                          const float* __restrict__ dinv,
                          const float* __restrict__ base, int n) {
    int idx = blockIdx.x * blockDim.x + threadIdx.x;   // one float4 per thread
    if (idx >= n * (HIDC / 4)) return;
    int node = idx >> 5;            // 32 float4 per node
    int f    = (idx & 31) * 4;
    float d  = dinv[node];
    float s  = d * d;
    const float4 v = *(const float4*)(S + (size_t)node * HIDC + f);
    float4 o;
    o.x = s * v.x; o.y = s * v.y; o.z = s * v.z; o.w = s * v.w;
    if (base) {
        const float4 b4 = *(const float4*)(base + (size_t)node * HIDC + f);
        o.x += b4.x; o.y += b4.y; o.z += b4.z; o.w += b4.w;
    }
    *(float4*)(T + (size_t)node * HIDC + f) = o;
}

// ---------------------------------------------------------------------------
// SpMM pass, step 2: edge scatter. One wave per edge; each lane owns a float4
// of the 128 features. Gather is a coalesced 512B burst from S[row]; scatter
// uses native f32 atomics (L2-resident: h is 25.6 MB << 192 MB L2).
// ---------------------------------------------------------------------------
__global__ void spmm_edges(float* __restrict__ T, const float* __restrict__ S,
                           const float* __restrict__ dinv,
                           const int* __restrict__ rows, const int* __restrict__ cols,
                           int nE) {
    int e = blockIdx.x * 8 + (threadIdx.x >> 5);   // blockDim = 256 -> 8 waves
    if (e >= nE) return;
    int lane = threadIdx.x & 31;
    int r = rows[e];
    int c = cols[e];
    float nm = dinv[r] * dinv[c];
    const float4 v = *(const float4*)(S + (size_t)r * HIDC + lane * 4);
    float* t = T + (size_t)c * HIDC + lane * 4;
    atomic_fadd(t + 0, nm * v.x);
    atomic_fadd(t + 1, nm * v.y);
    atomic_fadd(t + 2, nm * v.z);
    atomic_fadd(t + 3, nm * v.w);
}

// ---------------------------------------------------------------------------
// Host-side orchestration
// ---------------------------------------------------------------------------
extern "C" void kernel_launch(void* const* d_in, const int* in_sizes, int n_in,
                              void* d_out, int out_size, void* d_ws, size_t ws_size,
                              hipStream_t stream) {
    const float* x    = (const float*)d_in[0];
    const int*   ei   = (const int*)d_in[1];
    const int*   kg   = (const int*)d_in[2];
    const float* W_s1 = (const float*)d_in[3];
    const float* b_s1 = (const float*)d_in[4];
    const float* W_s2 = (const float*)d_in[5];
    const float* b_s2 = (const float*)d_in[6];
    const float* W_f2 = (const float*)d_in[7];
    const float* b_f2 = (const float*)d_in[8];

    const int hid = in_sizes[4];            // 128
    const int fin = in_sizes[3] / hid;      // 256
    const int N   = in_sizes[0] / fin;      // 50000
    const int Es  = in_sizes[1] / 2;        // 800000
    const int Ek  = in_sizes[2] / 2;        // 500000

    const int* rowS = ei;       const int* colS = ei + Es;
    const int* rowK = kg;       const int* colK = kg + Ek;

    // output segments: h0, h1, z0, z1
    size_t seg = (size_t)N * hid;
    float* out_h0 = (float*)d_out;
    float* out_h1 = out_h0 + seg;
    float* out_z0 = out_h0 + 2 * seg;
    float* out_z1 = out_h0 + 3 * seg;

    // workspace carve-up
    float*    bufA  = (float*)d_ws;              // N*128
    float*    bufB  = bufA + seg;                // N*128
    float*    dinvS = bufB + seg;                // N
    float*    dinvK = dinvS + N;                 // N
    unsigned* degS  = (unsigned*)(dinvK + N);    // N
    unsigned* degK  = degS + N;                  // N

    const int T = 256;
    dim3 blk(T);
    auto cdiv = [](int a, int b) { return (a + b - 1) / b; };

    // --- GCN normalization for both graphs ---
    fill_u32<<<cdiv(N, T), blk, 0, stream>>>(degS, 1u, N);   // self loop
    fill_u32<<<cdiv(N, T), blk, 0, stream>>>(degK, 1u, N);
    deg_count<<<cdiv(Es, T), blk, 0, stream>>>(degS, colS, Es);
    deg_count<<<cdiv(Ek, T), blk, 0, stream>>>(degK, colK, Ek);
    deg_to_dinv<<<cdiv(N, T), blk, 0, stream>>>(dinvS, degS, N);
    deg_to_dinv<<<cdiv(N, T), blk, 0, stream>>>(dinvK, degK, N);

    const int mTiles = N / 16;                   // 3125
    dim3 gGemm(cdiv(mTiles, 8)), bGemm(256);
    const int initGrid = cdiv(N * (hid / 4), T);

    // K-step propagation: src -> ... -> finalDst. Never writes into `src`
    // before its last read; for K==2 `src` itself is left untouched.
    auto prop = [&](float* finalDst, float* src, float* scratch, const float* dinv,
                    const int* rows, const int* cols, int nE, int K) {
        float* cur = src;
        float* other = scratch;
        for (int k = 0; k < K; ++k) {
            float* dst = (k == K - 1) ? finalDst : other;
            spmm_init<<<initGrid, blk, 0, stream>>>(dst, cur, dinv, nullptr, N);
            spmm_edges<<<cdiv(nE, 8), blk, 0, stream>>>(dst, cur, dinv, rows, cols, nE);
            other = cur;
            cur = dst;
        }
    };

    // --- h0 = prop^2(x @ W_s1^T + b_s1) ---
    gemm_xWt_bias<<<gGemm, bGemm, 0, stream>>>(x, W_s1, b_s1, bufA, mTiles);
    prop(out_h0, bufA, bufB, dinvS, rowS, colS, Es, 2);

    // --- h1 = prop^10(x @ W_s2^T + b_s2) ---
    gemm_xWt_bias<<<gGemm, bGemm, 0, stream>>>(x, W_s2, b_s2, bufA, mTiles);
    prop(out_h1, bufA, bufB, dinvS, rowS, colS, Es, 10);

    // --- z0 = prop^2(x @ W_f2^T + b_f2) ; bufA (dense f2) survives ---
    gemm_xWt_bias<<<gGemm, bGemm, 0, stream>>>(x, W_f2, b_f2, bufA, mTiles);
    prop(out_z0, bufA, bufB, dinvS, rowS, colS, Es, 2);

    // --- z1 = z0 + prop_knn^1(dense f2) ---
    spmm_init<<<initGrid, blk, 0, stream>>>(out_z1, bufA, dinvK, out_z0, N);
    spmm_edges<<<cdiv(Ek, 8), blk, 0, stream>>>(out_z1, bufA, dinvK, rowK, colK, Ek);
}